// MultiScaleDeformableAttention_52733608460641
// MI455X (gfx1250) — compile-verified
//
#include <hip/hip_runtime.h>
#include <hip/hip_bf16.h>
#include <math.h>

typedef __attribute__((ext_vector_type(2))) float v2f;
typedef __attribute__((ext_vector_type(4))) float v4f;
typedef __attribute__((ext_vector_type(8))) float v8f;

// ---- static problem config (matches reference) ----
#define BB        2
#define NQ        13294
#define NK        13294
#define DM        256
#define NH        8
#define NL        4
#define NP        4
#define CH        32              // == wave32 lanes
#define MROWS     (BB * NQ)       // 26588

__device__ __constant__ int   c_lvl_base[NL] = {0, 10000, 12500, 13125};
__device__ __constant__ int   c_H[NL] = {100, 50, 25, 13};
__device__ __constant__ int   c_W[NL] = {100, 50, 25, 13};

// ============================================================
// GEMM+bias:  Y[MROWS,N] = X[MROWS,256] * W[256,N] + bias[N]
// fp32 WMMA 16x16x4, one 16x16 tile per wave, 4 waves / block
// covering 64 rows x 16 cols.
// W col-tile (256x16) staged in LDS via GLOBAL_LOAD_ASYNC_TO_LDS
// (memory -> LDS direct, ASYNCcnt tracked, no VGPR round-trip).
// ============================================================
template <int N>
__global__ __launch_bounds__(128) void gemm_bias_wmma(
    const float* __restrict__ X, const float* __restrict__ W,
    const float* __restrict__ bias, float* __restrict__ Y)
{
    constexpr int TK = 256;
    // plain [k][n] layout: k-row = 16 floats = 64B contiguous
    __shared__ float ldsW[TK * 16];

    const int tid = threadIdx.x;
    const int n0  = blockIdx.y * 16;

    // ---- async stage W tile: 4096 floats = 1024 x b128 groups ----
    // group g -> floats [g*4 .. g*4+3]; k = g>>2, n = (g&3)*4
    // global row (k, n0..n0+15) is contiguous, so each 16B group is
    // contiguous in both global and LDS.
    {
        const float* wtile = W + n0;
        #pragma unroll
        for (int i = 0; i < 8; ++i) {
            const int g = tid + i * 128;
            const int k = g >> 2;
            const int n = (g & 3) * 4;
            const unsigned voff   = (unsigned)((k * N + n) * 4);
            const unsigned ldsoff = (unsigned)(size_t)(&ldsW[g * 4]);
            asm volatile("global_load_async_to_lds_b128 %0, %1, %2"
                         :: "v"(ldsoff), "v"(voff), "s"(wtile)
                         : "memory");
        }
        asm volatile("s_wait_asynccnt 0x0" ::: "memory");
    }
    __syncthreads();

    const int lane = tid & 31;
    const int wv   = tid >> 5;
    const int rlo  = lane & 15;       // M (for A) / N (for B,C,D) position
    const int hi   = lane >> 4;       // K-half selector
    const int m0   = blockIdx.x * 64 + wv * 16;

    int arow = m0 + rlo;
    if (arow >= MROWS) arow = MROWS - 1;          // clamp: EXEC must stay full
    const float* xrow = X + (size_t)arow * TK + 2 * hi;

    v8f c0 = {0.f, 0.f, 0.f, 0.f, 0.f, 0.f, 0.f, 0.f};
    v8f c1 = {0.f, 0.f, 0.f, 0.f, 0.f, 0.f, 0.f, 0.f};

    #pragma unroll 4
    for (int k0 = 0; k0 < TK; k0 += 8) {
        v2f a0 = *(const v2f*)(xrow + k0);
        v2f a1 = *(const v2f*)(xrow + k0 + 4);
        const int bb = (k0 + 2 * hi) * 16 + rlo;   // element (k, rlo)
        v2f b0 = { ldsW[bb],      ldsW[bb + 16] };      // k, k+1
        v2f b1 = { ldsW[bb + 64], ldsW[bb + 80] };      // k+4, k+5
        c0 = __builtin_amdgcn_wmma_f32_16x16x4_f32(false, a0, false, b0,
                                                   (short)0, c0, false, false);
        c1 = __builtin_amdgcn_wmma_f32_16x16x4_f32(false, a1, false, b1,
                                                   (short)0, c1, false, false);
    }

    const float bval = bias[n0 + rlo];
    float* yc = Y + (size_t)(m0 + 8 * hi) * N + n0 + rlo;
    if (m0 + 16 <= MROWS) {
        // full tile: straight-line coalesced stores (16 lanes = 64B / row)
        #pragma unroll
        for (int r = 0; r < 8; ++r)
            yc[(size_t)r * N] = c0[r] + c1[r] + bval;
    } else {
        #pragma unroll
        for (int r = 0; r < 8; ++r) {
            if (m0 + r + 8 * hi < MROWS)
                yc[(size_t)r * N] = c0[r] + c1[r] + bval;
        }
    }
}

// ============================================================
// Deformable sampling: one wave per (b, q, h); lane = channel.
// Fuses softmax over the 16 (level,point) logits.
// ============================================================
__global__ __launch_bounds__(256) void msda_sample(
    const float* __restrict__ value,   // (B*NK, 256) = (B,NK,H,CH)
    const float* __restrict__ offs,    // (B*NQ, 256) = (B,NQ,H,L,P,2)
    const float* __restrict__ attl,    // (B*NQ, 128) raw logits
    const float* __restrict__ refp,    // (B*NQ, L, 2)
    float* __restrict__ accb)          // (B*NQ, 256)
{
    const int lane = threadIdx.x & 31;
    const int wid  = blockIdx.x * (blockDim.x >> 5) + (threadIdx.x >> 5);
    const int total = BB * NQ * NH;
    if (wid >= total) return;

    const int h  = wid % NH;
    const int bq = wid / NH;          // b*NQ + q
    const int b  = bq / NQ;

    // ---- softmax over 16 logits (lane-redundant, 4x b128 loads) ----
    const float* lg = attl + (size_t)bq * (NH * NL * NP) + h * (NL * NP);
    float w[16];
    {
        v4f l0 = *(const v4f*)(lg + 0);
        v4f l1 = *(const v4f*)(lg + 4);
        v4f l2 = *(const v4f*)(lg + 8);
        v4f l3 = *(const v4f*)(lg + 12);
        #pragma unroll
        for (int j = 0; j < 4; ++j) { w[j] = l0[j]; w[4+j] = l1[j]; w[8+j] = l2[j]; w[12+j] = l3[j]; }
        float mx = w[0];
        #pragma unroll
        for (int j = 1; j < 16; ++j) mx = fmaxf(mx, w[j]);
        float s = 0.f;
        #pragma unroll
        for (int j = 0; j < 16; ++j) { w[j] = __expf(w[j] - mx); s += w[j]; }
        float inv = __frcp_rn(s);
        #pragma unroll
        for (int j = 0; j < 16; ++j) w[j] *= inv;
    }

    const float* orow = offs + (size_t)bq * DM + h * (NL * NP * 2);
    const float* rp   = refp + (size_t)bq * (NL * 2);

    float acc = 0.f;
    #pragma unroll
    for (int l = 0; l < NL; ++l) {
        const int   Hi = c_H[l], Wi = c_W[l];
        const float Hf = (float)Hi, Wf = (float)Wi;
        const float rx = rp[l * 2 + 0];
        const float ry = rp[l * 2 + 1];
        const float* vb = value + ((size_t)b * NK + c_lvl_base[l]) * DM + h * CH + lane;
        #pragma unroll
        for (int p = 0; p < NP; ++p) {
            const float ox = orow[(l * NP + p) * 2 + 0];
            const float oy = orow[(l * NP + p) * 2 + 1];
            // faithful quirk: normalizer is (H, W) -> x-offset / H, y-offset / W
            const float lx = rx + ox / Hf;
            const float ly = ry + oy / Wf;
            // grid = 2*loc-1; unnormalize (align_corners=False): x = loc.x*W - 0.5
            const float x = lx * Wf - 0.5f;
            const float y = ly * Hf - 0.5f;
            const float x0f = floorf(x), y0f = floorf(y);
            const float dx = x - x0f, dy = y - y0f;
            const int x0 = (int)x0f, y0 = (int)y0f;

            float s = 0.f;
            #pragma unroll
            for (int cy = 0; cy < 2; ++cy) {
                #pragma unroll
                for (int cx = 0; cx < 2; ++cx) {
                    const int xi = x0 + cx, yi = y0 + cy;
                    const bool ok = (xi >= 0) & (xi < Wi) & (yi >= 0) & (yi < Hi);
                    const int xc = min(max(xi, 0), Wi - 1);
                    const int yc = min(max(yi, 0), Hi - 1);
                    const float wt = (cx ? dx : 1.f - dx) * (cy ? dy : 1.f - dy);
                    const float v = vb[(size_t)(yc * Wi + xc) * DM];
                    s += ok ? v * wt : 0.f;
                }
            }
            acc += w[l * NP + p] * s;
        }
    }
    accb[(size_t)bq * DM + h * CH + lane] = acc;
}

// ============================================================
// host-side launch
// ============================================================
extern "C" void kernel_launch(void* const* d_in, const int* in_sizes, int n_in,
                              void* d_out, int out_size, void* d_ws, size_t ws_size,
                              hipStream_t stream) {
    const float* query  = (const float*)d_in[0];
    const float* refp   = (const float*)d_in[1];
    const float* inflat = (const float*)d_in[2];
    const float* Wv     = (const float*)d_in[3];
    const float* bv     = (const float*)d_in[4];
    const float* Ws     = (const float*)d_in[5];
    const float* bs     = (const float*)d_in[6];
    const float* Wa     = (const float*)d_in[7];
    const float* ba     = (const float*)d_in[8];
    const float* Wo     = (const float*)d_in[9];
    const float* bo     = (const float*)d_in[10];
    float* out = (float*)d_out;
    float* ws  = (float*)d_ws;

    // workspace layout (floats)
    float* value = ws;                                    // MROWS*256
    float* offs  = ws + (size_t)MROWS * 256;              // MROWS*256
    float* attl  = ws + (size_t)MROWS * 512;              // MROWS*128
    float* accb  = ws + (size_t)MROWS * 640;              // MROWS*256

    const int gx = (MROWS + 63) / 64;                     // 416
    dim3 blk(128);

    // 1) value = input_flatten @ Wv + bv
    gemm_bias_wmma<256><<<dim3(gx, 16), blk, 0, stream>>>(inflat, Wv, bv, value);
    // 2) raw sampling offsets = query @ Ws + bs
    gemm_bias_wmma<256><<<dim3(gx, 16), blk, 0, stream>>>(query, Ws, bs, offs);
    // 3) raw attention logits = query @ Wa + ba
    gemm_bias_wmma<128><<<dim3(gx, 8), blk, 0, stream>>>(query, Wa, ba, attl);
    // 4) softmax + deformable bilinear sampling -> accb
    {
        const int totalWaves = BB * NQ * NH;              // 212704, exactly 8 per block
        const int blocks = (totalWaves + 7) / 8;          // 26588
        msda_sample<<<blocks, 256, 0, stream>>>(value, offs, attl, refp, accb);
    }
    // 5) out = accb @ Wo + bo
    gemm_bias_wmma<256><<<dim3(gx, 16), blk, 0, stream>>>(accb, Wo, bo, out);
}